// SSIM_loss_52441550684315
// MI455X (gfx1250) — compile-verified
//
#include <hip/hip_runtime.h>
#include <hip/hip_bf16.h>

// SSIM loss, fused single-pass per 16-output-row band.
// - Global -> LDS staging via CDNA5 async copy (global_load_async_to_lds_b128,
//   ASYNCcnt-tracked, VGPR-bypassing), 16B per lane per op.
// - Vertical 7-tap box filter done as GEMM with banded constant A via
//   v_wmma_f32_16x16x4_f32 (K=22 padded to 24 -> 6 chained WMMAs).

typedef __attribute__((ext_vector_type(2))) float v2f;
typedef __attribute__((ext_vector_type(8))) float v8f;
typedef __attribute__((address_space(3))) float lds_f;

#define Hh   256
#define Ww   256
#define OH   250
#define OW   250
#define BAND 16          // output rows per workgroup
#define INROWS 22        // input rows needed per band (16 + 6)
#define KPAD 24          // K padded to multiple of 4
#define LSTR 260         // LDS row stride (260 % 64 == 4 -> de-aliased banks)
#define NSLICE 256       // 4 * 64
#define NBAND 16         // ceil(250/16)

__device__ __forceinline__ unsigned lds_off(const float* p) {
    // generic -> LDS addrspace cast; low 32 bits are the workgroup-relative
    // LDS byte address used by the async-copy VDST operand.
    return (unsigned)(unsigned long long)(lds_f*)p;
}

__global__ __launch_bounds__(256, 1)
void ssim_band_kernel(const float* __restrict__ X,
                      const float* __restrict__ Y,
                      float* __restrict__ partial) {
    __shared__ float xs[INROWS * LSTR];
    __shared__ float ys[INROWS * LSTR];
    __shared__ float hs[5][KPAD * LSTR];
    __shared__ float red[8];

    const int band  = blockIdx.x;          // 0..15
    const int slice = blockIdx.y;          // 0..255
    const int r0    = band * BAND;         // first output row of band
    const int tid   = threadIdx.x;         // 256 threads = 8 waves
    const long base = (long)slice * (Hh * Ww);

    // ---- Stage 1: async-copy 22x256 band of X and Y into LDS (B128 per lane).
    // 22 rows x 64 float4 = 1408 vec4 elements per input.
    for (int idx = tid; idx < INROWS * (Ww / 4); idx += 256) {
        const int r  = idx >> 6;           // 0..21
        const int c4 = idx & 63;           // float4 column
        const int gr = r0 + r;
        const unsigned lx = lds_off(&xs[r * LSTR + c4 * 4]);
        const unsigned ly = lds_off(&ys[r * LSTR + c4 * 4]);
        if (gr < Hh) {
            const unsigned voff =
                (unsigned)((base + (long)gr * Ww + c4 * 4) * sizeof(float));
            asm volatile("global_load_async_to_lds_b128 %0, %1, %2"
                         :: "v"(lx), "v"(voff), "s"(X) : "memory");
            asm volatile("global_load_async_to_lds_b128 %0, %1, %2"
                         :: "v"(ly), "v"(voff), "s"(Y) : "memory");
        } else {
            // zero-pad rows past the image (only last band)
            *(float4*)&xs[r * LSTR + c4 * 4] = make_float4(0.f, 0.f, 0.f, 0.f);
            *(float4*)&ys[r * LSTR + c4 * 4] = make_float4(0.f, 0.f, 0.f, 0.f);
        }
    }
    asm volatile("s_wait_asynccnt 0x0" ::: "memory");
    __syncthreads();

    // ---- Stage 2: horizontal 7-tap sums for the 5 maps.
    // Rows 22..23 and cols >= 250 written as true zeros (K-padding for WMMA;
    // WMMA propagates NaN even when A is 0, so padding must be clean).
    for (int idx = tid; idx < KPAD * Ww; idx += 256) {
        const int r = idx >> 8;            // 0..23
        const int c = idx & 255;
        float s0 = 0.f, s1 = 0.f, s2 = 0.f, s3 = 0.f, s4 = 0.f;
        if (r < INROWS && c < OW) {
            #pragma unroll
            for (int j = 0; j < 7; ++j) {
                const float xv = xs[r * LSTR + c + j];
                const float yv = ys[r * LSTR + c + j];
                s0 += xv;       s1 += yv;
                s2 += xv * xv;  s3 += yv * yv;  s4 += xv * yv;
            }
        }
        hs[0][r * LSTR + c] = s0;
        hs[1][r * LSTR + c] = s1;
        hs[2][r * LSTR + c] = s2;
        hs[3][r * LSTR + c] = s3;
        hs[4][r * LSTR + c] = s4;
    }
    __syncthreads();

    // ---- Stage 3: vertical 7-tap as GEMM. out = Wv(16x24) x hsum(24x16cols)
    const int wave  = tid >> 5;
    const int lane  = tid & 31;
    const int lhalf = lane >> 4;           // 0: lanes 0-15, 1: lanes 16-31
    const int lm    = lane & 15;

    // Banded A per documented f32 16x4 A layout:
    //   lanes 0-15:  M=lane,    VGPRv -> K = 4q + v
    //   lanes 16-31: M=lane-16, VGPRv -> K = 4q + 2 + v
    v2f Aq[6];
    #pragma unroll
    for (int q = 0; q < 6; ++q) {
        #pragma unroll
        for (int v = 0; v < 2; ++v) {
            const int K = 4 * q + 2 * lhalf + v;
            Aq[q][v] = (K >= lm && K <= lm + 6) ? 1.0f : 0.0f;
        }
    }

    const float inv49 = 1.0f / 49.0f;
    const float covn  = 49.0f / 48.0f;
    const float C1    = 1.0e-4f;           // (0.01*1.0)^2
    const float C2    = 9.0e-4f;           // (0.03*1.0)^2

    float lsum = 0.0f;

    #pragma unroll
    for (int t = 0; t < 2; ++t) {
        const int ct = wave + 8 * t;       // column tile 0..15
        const int c0 = ct * 16;

        v8f acc[5];
        #pragma unroll
        for (int p = 0; p < 5; ++p) {
            v8f c = {0.f, 0.f, 0.f, 0.f, 0.f, 0.f, 0.f, 0.f};
            #pragma unroll
            for (int q = 0; q < 6; ++q) {
                // B (4x16) dual layout: lanes 0-15 hold K = 4q+v, lanes
                // 16-31 hold K = 4q+2+v, column N = lane%16.
                const int rb = 4 * q + 2 * lhalf;
                v2f b;
                b[0] = hs[p][(rb + 0) * LSTR + c0 + lm];
                b[1] = hs[p][(rb + 1) * LSTR + c0 + lm];
                c = __builtin_amdgcn_wmma_f32_16x16x4_f32(
                        false, Aq[q], false, b, (short)0, c, false, false);
            }
            acc[p] = c;
        }

        // ---- Stage 4: SSIM pointwise on D tiles.
        // C/D layout: element i -> M = i + 8*(lane>=16), N = lane%16
        #pragma unroll
        for (int i = 0; i < 8; ++i) {
            const int m    = i + 8 * lhalf;
            const int grow = r0 + m;
            const int gcol = c0 + lm;
            if (grow < OH && gcol < OW) {
                const float ux  = acc[0][i] * inv49;
                const float uy  = acc[1][i] * inv49;
                const float uxx = acc[2][i] * inv49;
                const float uyy = acc[3][i] * inv49;
                const float uxy = acc[4][i] * inv49;
                const float vx  = covn * (uxx - ux * ux);
                const float vy  = covn * (uyy - uy * uy);
                const float vxy = covn * (uxy - ux * uy);
                const float A1  = 2.0f * ux * uy + C1;
                const float A2  = 2.0f * vxy + C2;
                const float B1  = ux * ux + uy * uy + C1;
                const float B2  = vx + vy + C2;
                lsum += (A1 * A2) / (B1 * B2);
            }
        }
    }

    // ---- Stage 5: deterministic reduction -> one partial per workgroup.
    #pragma unroll
    for (int off = 16; off >= 1; off >>= 1)
        lsum += __shfl_xor(lsum, off, 32);
    if (lane == 0) red[wave] = lsum;
    __syncthreads();
    if (tid == 0) {
        float s = 0.0f;
        #pragma unroll
        for (int k = 0; k < 8; ++k) s += red[k];
        partial[blockIdx.y * NBAND + blockIdx.x] = s;
    }
}

__global__ __launch_bounds__(256, 1)
void ssim_finish_kernel(const float* __restrict__ partial,
                        float* __restrict__ out) {
    __shared__ float red[8];
    const int tid = threadIdx.x;
    float s = 0.0f;
    for (int i = tid; i < NSLICE * NBAND; i += 256) s += partial[i];
    #pragma unroll
    for (int off = 16; off >= 1; off >>= 1)
        s += __shfl_xor(s, off, 32);
    if ((tid & 31) == 0) red[tid >> 5] = s;
    __syncthreads();
    if (tid == 0) {
        float tot = 0.0f;
        #pragma unroll
        for (int k = 0; k < 8; ++k) tot += red[k];
        const float loss = 1.0f - tot / 16000000.0f; // 4*64*250*250
        out[0] = loss; out[1] = loss; out[2] = loss; out[3] = loss;
    }
}

extern "C" void kernel_launch(void* const* d_in, const int* in_sizes, int n_in,
                              void* d_out, int out_size, void* d_ws, size_t ws_size,
                              hipStream_t stream) {
    (void)in_sizes; (void)n_in; (void)out_size; (void)ws_size;
    const float* X = (const float*)d_in[0];
    const float* Y = (const float*)d_in[1];
    // d_in[2] is the box-filter weight buffer (constant 1/49), not needed.
    float* partial = (float*)d_ws;                 // 4096 floats
    float* out     = (float*)d_out;                // 4 floats

    dim3 grid(NBAND, NSLICE);
    ssim_band_kernel<<<grid, 256, 0, stream>>>(X, Y, partial);
    ssim_finish_kernel<<<1, 256, 0, stream>>>(partial, out);
}